// Graph_Encoder_53584011985720
// MI455X (gfx1250) — compile-verified
//
#include <hip/hip_runtime.h>
#include <hip/hip_bf16.h>

#define DIM 256
#define BN_EPS 1e-5f
#define LDA 260       // padded LDS row stride (floats): kills 64-bank conflicts, 1040B = 65*16 stays float4-aligned
#define MT 5          // M tiles (of 16 rows) per block
#define ROWS (MT*16)  // 80 rows per block; 100000/80 = 1250 exact

typedef __attribute__((ext_vector_type(2))) float v2f;
typedef __attribute__((ext_vector_type(8))) float v8f;
typedef __attribute__((ext_vector_type(4))) unsigned int u32x4;
typedef __attribute__((ext_vector_type(8))) int i32x8;
typedef __attribute__((ext_vector_type(4))) int i32x4;

// ---------------- init kernels ----------------
__global__ __launch_bounds__(256) void k_init_deg(float* __restrict__ deg, int n) {
    int i = blockIdx.x * 256 + threadIdx.x;
    if (i < n) deg[i] = 1.0f;  // self-loop contributes 1 to every degree
}

__global__ __launch_bounds__(256) void k_degree(const int* __restrict__ col,
                                                float* __restrict__ deg, int nE) {
    int e = blockIdx.x * 256 + threadIdx.x;
    if (e < nE) atomicAdd(&deg[col[e]], 1.0f);
}

__global__ __launch_bounds__(256) void k_rsqrt_inplace(float* __restrict__ deg, int n) {
    int i = blockIdx.x * 256 + threadIdx.x;
    if (i < n) deg[i] = rsqrtf(deg[i]);   // deg >= 1 always
}

// ---------------- fp32 WMMA GEMM: h = x @ W ----------------
// block: 256 threads = 8 waves. 80 rows (5 M-tiles) per block; wave w owns columns [32w, 32w+32).
// One B-fragment load feeds 10 WMMAs (5 M-tiles x 2 N-subtiles) -> 5x less W traffic from L2.
__global__ __launch_bounds__(256) void k_gemm_wmma(const float* __restrict__ x,
                                                   const float* __restrict__ W,
                                                   float* __restrict__ h) {
    __shared__ float sA[ROWS * LDA];
    const int tid  = threadIdx.x;
    const int row0 = blockIdx.x * ROWS;

#if __has_builtin(__builtin_amdgcn_tensor_load_to_lds) && __has_builtin(__builtin_amdgcn_s_wait_tensorcnt)
    // ---- TDM path: one DMA of the 80x256 f32 tile, HW-padded to a 260-float LDS row stride ----
    if (tid < 32) {                         // wave 0 issues the tensor DMA (TDM ignores EXEC; gate by wave)
        const unsigned ldsOff = (unsigned)(uintptr_t)sA;   // generic LDS ptr low 32 bits = LDS byte offset
        const unsigned long long ga = (unsigned long long)(const void*)(x + (size_t)row0 * DIM);
        u32x4 g0;
        g0[0] = 1u;                                            // count=1, user descriptor
        g0[1] = ldsOff;                                        // lds_addr (bytes)
        g0[2] = (unsigned)ga;                                  // global_addr[31:0]
        g0[3] = ((unsigned)(ga >> 32) & 0x01FFFFFFu)           // global_addr[56:32]
              | (2u << 30);                                    // type = 2 ("image")
        i32x8 g1;
        g1[0] = (int)((2u << 16)        // data_size = 4 bytes
              |       (1u << 20)        // pad_enable
              |       (7u << 22)        // pad_interval: every 256 DWORDs (one 256-float row)
              |       (3u << 25));      // pad_amount: 4 DWORDs -> LDS row stride 260 floats
        g1[1] = (int)(256u << 16);      // tensor_dim0 = 256 (lo 16)
        g1[2] = (int)((unsigned)ROWS << 16);  // tensor_dim0 hi = 0 | tensor_dim1 = 80
        g1[3] = (int)(256u << 16);      // tensor_dim1 hi = 0 | tile_dim0 = 256
        g1[4] = ROWS;                   // tile_dim1 = 80 | tile_dim2 = 0
        g1[5] = 256;                    // tensor_dim0_stride[31:0] = 256 elements
        g1[6] = 0;                      // stride hi | tensor_dim1_stride lo (unused, 2-D)
        g1[7] = 0;
        const i32x4 gz4 = {0, 0, 0, 0};             // groups 2/3 unused for 2-D tile
        const i32x8 gz8 = {0, 0, 0, 0, 0, 0, 0, 0}; // 6-arg (clang-23) form: extra group slot
        __builtin_amdgcn_tensor_load_to_lds(g0, g1, gz4, gz4, gz8, 0);
        __builtin_amdgcn_s_wait_tensorcnt(0);
    }
    __syncthreads();
#else
    // ---- fallback: coalesced float4 copy with manual padding ----
    {
        const float4* src = (const float4*)(x + (size_t)row0 * DIM);
        for (int idx = tid; idx < ROWS * 64; idx += 256) {
            float4 v = src[idx];
            int r = idx >> 6;             // 64 float4 per 256-float row
            int c = (idx & 63) << 2;
            *(float4*)&sA[r * LDA + c] = v;
        }
    }
    __syncthreads();
#endif

    const int wave  = tid >> 5;       // 0..7 (wave32)
    const int lane  = tid & 31;
    const int ncol0 = wave * 32;

    // ISA 16x16x4 f32 operand layout:
    //   A: lane<16 holds (M=lane, K=k+0..1), lane>=16 holds (M=lane-16, K=k+2..3)
    //   B: lane<16 holds (N=lane, K=k+0..1), lane>=16 holds (N=lane-16, K=k+2..3)
    const int half = lane >> 4;       // 0 or 1
    const int sub  = lane & 15;
    const int kOff = half * 2;

    v8f acc[MT][2] = {};

    for (int k = 0; k < DIM; k += 4) {
        const float* Wk = W + (size_t)(k + kOff) * DIM + ncol0 + sub;
        v2f b0, b1;
        b0.x = Wk[0];        b0.y = Wk[DIM];         // columns ncol0 .. ncol0+15
        b1.x = Wk[16];       b1.y = Wk[DIM + 16];    // columns ncol0+16 .. ncol0+31
        #pragma unroll
        for (int t = 0; t < MT; ++t) {
            v2f a = *(const v2f*)&sA[(t * 16 + sub) * LDA + k + kOff];
            acc[t][0] = __builtin_amdgcn_wmma_f32_16x16x4_f32(false, a, false, b0,
                                                              (short)0, acc[t][0], false, false);
            acc[t][1] = __builtin_amdgcn_wmma_f32_16x16x4_f32(false, a, false, b1,
                                                              (short)0, acc[t][1], false, false);
        }
    }

    // C/D layout: VGPR r, lane<16 -> (M=r, N=lane); lane>=16 -> (M=r+8, N=lane-16)
    const int mBase = half * 8;
    #pragma unroll
    for (int t = 0; t < MT; ++t) {
        float* dst = h + (size_t)(row0 + t * 16 + mBase) * DIM + ncol0 + sub;
        #pragma unroll
        for (int r = 0; r < 8; ++r) {
            dst[(size_t)r * DIM]      = acc[t][0][r];
            dst[(size_t)r * DIM + 16] = acc[t][1][r];
        }
    }
}

// ---------------- edge scatter: acc[i] += h[j] * dinv[i]*dinv[j] ----------------
__global__ __launch_bounds__(256) void k_scatter(const float* __restrict__ h,
                                                 const int* __restrict__ row,
                                                 const int* __restrict__ col,
                                                 const float* __restrict__ dinv,
                                                 float* __restrict__ acc) {
    const int e = blockIdx.x;          // one edge per block
    const int f = threadIdx.x;         // feature
    const int j = row[e];              // source (uniform -> scalar load)
    const int i = col[e];              // target
    const float nrm = dinv[i] * dinv[j];
    const float v = h[(size_t)j * DIM + f] * nrm;
    atomicAdd(&acc[(size_t)i * DIM + f], v);
}

// ---------------- self-loop + bias, fused with BN statistics ----------------
__global__ __launch_bounds__(256) void k_selfloop_stats(const float* __restrict__ h,
                                                        const float* __restrict__ dinv,
                                                        const float* __restrict__ b,
                                                        float* __restrict__ acc,
                                                        float* __restrict__ sums,
                                                        int nodesPerBlock, int nNodes) {
    const int f  = threadIdx.x;
    const int n0 = blockIdx.x * nodesPerBlock;
    const int n1 = min(n0 + nodesPerBlock, nNodes);
    const float bf = b[f];
    float s = 0.0f, s2 = 0.0f;
    for (int n = n0; n < n1; ++n) {
        const float di = dinv[n];
        float val = acc[(size_t)n * DIM + f] + h[(size_t)n * DIM + f] * di * di + bf;
        acc[(size_t)n * DIM + f] = val;
        s  += val;
        s2 += val * val;
    }
    atomicAdd(&sums[f], s);
    atomicAdd(&sums[DIM + f], s2);
}

// ---------------- batch-norm + tanh (in place on d_out) ----------------
__global__ __launch_bounds__(256) void k_bn_tanh(float* __restrict__ out,
                                                 const float* __restrict__ sums,
                                                 const float* __restrict__ gamma,
                                                 const float* __restrict__ beta,
                                                 int nNodes) {
    const int f = threadIdx.x;
    const int n = blockIdx.x;
    const float invN = 1.0f / (float)nNodes;
    const float mean = sums[f] * invN;
    const float var  = sums[DIM + f] * invN - mean * mean;
    const float scale = rsqrtf(var + BN_EPS) * gamma[f];
    const float shift = beta[f];
    const size_t idx = (size_t)n * DIM + f;
    out[idx] = tanhf((out[idx] - mean) * scale + shift);
}

extern "C" void kernel_launch(void* const* d_in, const int* in_sizes, int n_in,
                              void* d_out, int out_size, void* d_ws, size_t ws_size,
                              hipStream_t stream) {
    const float* x     = (const float*)d_in[0];
    const float* W     = (const float*)d_in[1];
    const float* b     = (const float*)d_in[2];
    const float* gamma = (const float*)d_in[3];
    const float* beta  = (const float*)d_in[4];
    const int*   ei    = (const int*)d_in[5];

    const int nNodes = in_sizes[0] / DIM;     // 100000
    const int nE     = in_sizes[5] / 2;       // 320000
    const int* row = ei;                      // edge_index[0] = sources
    const int* col = ei + nE;                 // edge_index[1] = targets

    float* out  = (float*)d_out;                          // scatter accumulator, then final output
    float* h    = (float*)d_ws;                           // [nNodes*DIM] x @ W
    float* deg  = h + (size_t)nNodes * DIM;               // [nNodes] degree -> dinv (in place)
    float* sums = deg + nNodes;                           // [2*DIM] sum / sumsq per feature

    (void)hipMemsetAsync(d_out, 0, (size_t)nNodes * DIM * sizeof(float), stream);
    (void)hipMemsetAsync(sums, 0, 2 * DIM * sizeof(float), stream);

    k_init_deg     <<<(nNodes + 255) / 256, 256, 0, stream>>>(deg, nNodes);
    k_degree       <<<(nE + 255) / 256,     256, 0, stream>>>(col, deg, nE);
    k_rsqrt_inplace<<<(nNodes + 255) / 256, 256, 0, stream>>>(deg, nNodes);

    k_gemm_wmma    <<<nNodes / ROWS, 256, 0, stream>>>(x, W, h);   // 1250 blocks, exact

    k_scatter      <<<nE, 256, 0, stream>>>(h, row, col, deg, out);

    const int nodesPerBlock = 250;                                  // 400 blocks
    k_selfloop_stats<<<(nNodes + nodesPerBlock - 1) / nodesPerBlock, 256, 0, stream>>>(
        h, deg, b, out, sums, nodesPerBlock, nNodes);

    k_bn_tanh      <<<nNodes, 256, 0, stream>>>(out, sums, gamma, beta, nNodes);
}